// NodeAttention_37185826848828
// MI455X (gfx1250) — compile-verified
//
#include <hip/hip_runtime.h>
#include <math.h>

#define PDIM 6
#define HDIM 8
#define NDIM 1024
#define DDIM 64
#define K0   16
#define K1   8
#define D1   128          // H*K0
#define OUTC 256          // 64 + 128 + 64
#define LP   1028         // padded LDS row stride (floats): conflict-free, rows stay 16B-aligned

typedef float v2f __attribute__((ext_vector_type(2)));
typedef float v8f __attribute__((ext_vector_type(8)));

__device__ __forceinline__ v8f wmma_f32_4(v2f a, v2f b, v8f c) {
  // V_WMMA_F32_16X16X4_F32: D = A(16x4) * B(4x16) + C(16x16), full f32
  return __builtin_amdgcn_wmma_f32_16x16x4_f32(false, a, false, b, (short)0, c, false, false);
}

__device__ __forceinline__ float lrelu02(float v) { return v >= 0.f ? v : 0.2f * v; }

// Butterfly xor across lanes via a single DS_SWIZZLE_B32 (group-of-32 pattern:
// offset = {1'b0, xor[4:0], or[4:0]=0, and[4:0]=0x1f}).
template <int XM>
__device__ __forceinline__ float swz_xor(float v) {
  return __int_as_float(
      __builtin_amdgcn_ds_swizzle(__float_as_int(v), (XM << 10) | 0x1f));
}

// CDNA5 async memory->LDS copy (ASYNCcnt-tracked, no VGPR data path).
__device__ __forceinline__ void async_load_b128(unsigned lds_off, const void* gaddr) {
  asm volatile("global_load_async_to_lds_b128 %0, %1, off"
               :
               : "v"(lds_off), "v"(gaddr)
               : "memory");
}

__device__ __forceinline__ void async_wait0() {
#if __has_builtin(__builtin_amdgcn_s_wait_asynccnt)
  __builtin_amdgcn_s_wait_asynccnt(0);
#else
  asm volatile("s_wait_asynccnt 0x0" ::: "memory");
#endif
}

// ---------------------------------------------------------------------------
// out[n][p][0:64] = X[n][0:64]
__global__ void xcopy_kernel(const float* __restrict__ X, float* __restrict__ out) {
  int idx = blockIdx.x * blockDim.x + threadIdx.x;
  if (idx >= NDIM * PDIM * DDIM) return;
  int c = idx & (DDIM - 1);
  int t = idx / DDIM;
  int p = t % PDIM;
  int n = t / PDIM;
  out[(size_t)n * (PDIM * OUTC) + p * OUTC + c] = X[(size_t)n * DDIM + c];
}

// ---------------------------------------------------------------------------
// feats[p,h,n,k] = Xin @ W ;  s = feats·aS ; t = feats·aN
// one wave per (p,h,16-row tile); WMMA f32 16x16x4 over DIN/4 K-steps.
template <int DIN, int K, bool PERP>
__global__ void proj_kernel(const float* __restrict__ Xin, const float* __restrict__ W,
                            const float* __restrict__ aS, const float* __restrict__ aN,
                            float* __restrict__ feats, float* __restrict__ sOut,
                            float* __restrict__ tOut) {
  const int lane = threadIdx.x & 31;
  const int wave = threadIdx.x >> 5;
  const int wid = blockIdx.x * 8 + wave;
  const int ntiles = NDIM / 16;
  const int p = wid / (HDIM * ntiles);
  const int rem = wid % (HDIM * ntiles);
  const int h = rem / ntiles;
  const int n0 = (rem % ntiles) * 16;
  const int half = lane >> 4;     // K-halves: lanes 0-15 hold K={0,1}, 16-31 hold K={2,3}
  const int l16 = lane & 15;
  const int ph = p * HDIM + h;

  // A-operand: lane l16 holds matrix row M=l16
  const float* Xrow = PERP ? (Xin + ((size_t)p * NDIM + n0 + l16) * DIN)
                           : (Xin + (size_t)(n0 + l16) * DIN);
  const float* Wb = W + (size_t)ph * DIN * K;
  const int col = l16;            // B/D: lane l16 holds column N=l16
  const bool valid = (col < K);

  v8f c = {};
#pragma unroll 4
  for (int kk = 0; kk < DIN / 4; ++kk) {
    const int kbase = 4 * kk + 2 * half;
    const float2 av = *reinterpret_cast<const float2*>(Xrow + kbase);
    v2f a; a[0] = av.x; a[1] = av.y;
    v2f b;
    b[0] = valid ? Wb[(size_t)kbase * K + col] : 0.f;
    b[1] = valid ? Wb[(size_t)(kbase + 1) * K + col] : 0.f;
    c = wmma_f32_4(a, b, c);
  }

  const float aSv = valid ? aS[ph * K + col] : 0.f;
  const float aNv = valid ? aN[ph * K + col] : 0.f;

#pragma unroll
  for (int r = 0; r < 8; ++r) {
    const int row = n0 + r + half * 8;       // D layout: reg r = row r (lo half) / 8+r (hi half)
    const float fv = c[r];
    if (valid) feats[((size_t)ph * NDIM + row) * K + col] = fv;
    float sv = fv * aSv;
    float tv = fv * aNv;
    // reduce across the 16 lanes of each half (xor masks stay inside the half)
    sv += swz_xor<1>(sv);  tv += swz_xor<1>(tv);
    sv += swz_xor<2>(sv);  tv += swz_xor<2>(tv);
    sv += swz_xor<4>(sv);  tv += swz_xor<4>(tv);
    sv += swz_xor<8>(sv);  tv += swz_xor<8>(tv);
    if (l16 == 0) {
      sOut[(size_t)ph * NDIM + row] = sv;
      tOut[(size_t)ph * NDIM + row] = tv;
    }
  }
}

// ---------------------------------------------------------------------------
// One block = (p, 16-row tile). A tile async-staged ONCE into LDS, reused by
// all 8 head-waves (8x HBM traffic cut on the dominant stream). Per wave:
// two-pass softmax over the LDS tile, then WMMA-accumulated attn @ feats.
template <int K, bool WRITE_H1>
__global__ void attn_kernel(const float* __restrict__ A, const float* __restrict__ feats,
                            const float* __restrict__ sIn, const float* __restrict__ tIn,
                            const float* __restrict__ bias, float* __restrict__ h1,
                            float* __restrict__ out, int outOff) {
  extern __shared__ float smem[];
  float* Atile = smem;               // 16 * LP floats (padded rows, conflict-free)
  float* Tlds = smem + 16 * LP;      // HDIM * NDIM floats (per-head neighbor logits)

  const int tid = threadIdx.x;
  const int ntiles = NDIM / 16;
  const int p = blockIdx.x / ntiles;
  const int i0 = (blockIdx.x % ntiles) * 16;

  {  // async b128 staging: memory -> LDS direct, tracked by ASYNCcnt
    const char* Ag = (const char*)(A + ((size_t)p * NDIM + i0) * NDIM);
    for (int idx = tid; idx < 16 * (NDIM / 4); idx += blockDim.x) {
      const int r = idx >> 8;                       // NDIM/4 == 256
      const int c4 = idx & 255;
      const unsigned loff = (unsigned)(r * (LP * 4) + c4 * 16);
      async_load_b128(loff, Ag + ((size_t)r * NDIM + (size_t)c4 * 4) * sizeof(float));
    }
    const char* Tg = (const char*)(tIn + (size_t)p * HDIM * NDIM);
    const unsigned tbase = 16u * LP * 4u;
    for (int idx = tid; idx < HDIM * (NDIM / 4); idx += blockDim.x) {
      async_load_b128(tbase + (unsigned)idx * 16u, Tg + (size_t)idx * 16);
    }
    async_wait0();
  }
  __syncthreads();

  const int h = tid >> 5;            // wave == head
  const int lane = tid & 31;
  const int half = lane >> 4;
  const int l16 = lane & 15;
  const int ph = p * HDIM + h;
  const float* Trow = Tlds + h * NDIM;
  const float* Arow = Atile + l16 * LP;          // this lane's attention row
  const float s_i = sIn[(size_t)ph * NDIM + i0 + l16];

  // pass 1: row max then row sum; halves split the j-range, combined by swizzle
  const int jbase = half * (NDIM / 2);
  float m = -3.4e38f;
  for (int j = 0; j < NDIM / 2; ++j) {
    const int jj = jbase + j;
    m = fmaxf(m, lrelu02(s_i + Trow[jj]) + Arow[jj]);
  }
  m = fmaxf(m, swz_xor<16>(m));
  float l = 0.f;
  for (int j = 0; j < NDIM / 2; ++j) {
    const int jj = jbase + j;
    l += __expf(lrelu02(s_i + Trow[jj]) + Arow[jj] - m);
  }
  l += swz_xor<16>(l);
  const float inv_l = 1.f / l;

  // pass 2: D[i, k] = sum_j softmax(i,j) * feats[j, k] via WMMA f32 16x16x4
  const int col = l16;
  const bool valid = (col < K);
  const float* fH = feats + (size_t)ph * NDIM * K;
  v8f c = {};
  for (int kk = 0; kk < NDIM / 4; ++kk) {
    const int j = 4 * kk + 2 * half;             // lane's two K-slots
    if ((kk & 15) == 0 && kk + 16 < NDIM / 4)
      __builtin_prefetch(fH + (size_t)(4 * (kk + 16)) * K, 0, 1);  // global_prefetch_b8
    const float d0 = lrelu02(s_i + Trow[j]) + Arow[j];
    const float d1 = lrelu02(s_i + Trow[j + 1]) + Arow[j + 1];
    v2f a; a[0] = __expf(d0 - m) * inv_l; a[1] = __expf(d1 - m) * inv_l;
    v2f b;
    b[0] = valid ? fH[(size_t)j * K + col] : 0.f;
    b[1] = valid ? fH[(size_t)(j + 1) * K + col] : 0.f;
    c = wmma_f32_4(a, b, c);
  }

  if (!valid) return;
  const float bb = bias[ph * K + col];
#pragma unroll
  for (int r = 0; r < 8; ++r) {
    const int row = i0 + r + half * 8;
    float v = c[r] + bb;
    v = v > 0.f ? v : expm1f(v);                 // ELU
    if (WRITE_H1) h1[((size_t)p * NDIM + row) * D1 + h * K + col] = v;
    out[(size_t)row * (PDIM * OUTC) + p * OUTC + outOff + h * K + col] = v;
  }
}

// ---------------------------------------------------------------------------
extern "C" void kernel_launch(void* const* d_in, const int* in_sizes, int n_in,
                              void* d_out, int out_size, void* d_ws, size_t ws_size,
                              hipStream_t stream) {
  (void)in_sizes; (void)n_in; (void)out_size; (void)ws_size;
  const float* X   = (const float*)d_in[0];
  const float* A   = (const float*)d_in[1];
  const float* W0  = (const float*)d_in[2];
  const float* b0  = (const float*)d_in[3];
  const float* aS0 = (const float*)d_in[4];
  const float* aN0 = (const float*)d_in[5];
  const float* W1  = (const float*)d_in[6];
  const float* b1  = (const float*)d_in[7];
  const float* aS1 = (const float*)d_in[8];
  const float* aN1 = (const float*)d_in[9];
  float* out = (float*)d_out;
  float* ws = (float*)d_ws;

  float* feats0 = ws;                                           // P*H*N*K0
  float* s0 = feats0 + (size_t)PDIM * HDIM * NDIM * K0;         // P*H*N
  float* t0 = s0 + (size_t)PDIM * HDIM * NDIM;                  // P*H*N
  float* h1 = t0 + (size_t)PDIM * HDIM * NDIM;                  // P*N*D1
  float* feats1 = h1 + (size_t)PDIM * NDIM * D1;                // P*H*N*K1
  float* s1 = feats1 + (size_t)PDIM * HDIM * NDIM * K1;
  float* t1 = s1 + (size_t)PDIM * HDIM * NDIM;

  const dim3 blk(256);
  const size_t lds = (16 * LP + HDIM * NDIM) * sizeof(float);   // ~98.5 KB of 320 KB/WGP

  xcopy_kernel<<<(NDIM * PDIM * DDIM + 255) / 256, blk, 0, stream>>>(X, out);

  proj_kernel<DDIM, K0, false>
      <<<PDIM * HDIM * (NDIM / 16) / 8, blk, 0, stream>>>(X, W0, aS0, aN0, feats0, s0, t0);
  attn_kernel<K0, true>
      <<<PDIM * (NDIM / 16), blk, lds, stream>>>(A, feats0, s0, t0, b0, h1, out, DDIM);

  proj_kernel<D1, K1, true>
      <<<PDIM * HDIM * (NDIM / 16) / 8, blk, 0, stream>>>(h1, W1, aS1, aN1, feats1, s1, t1);
  attn_kernel<K1, false>
      <<<PDIM * (NDIM / 16), blk, lds, stream>>>(A, feats1, s1, t1, b1, nullptr, out,
                                                 DDIM + HDIM * K0);
}